// SimpleRNN_5420248727639
// MI455X (gfx1250) — compile-verified
//
#include <hip/hip_runtime.h>
#include <hip/hip_bf16.h>
#include <math.h>

// ---------------------------------------------------------------------------
// Two-layer tanh RNN on gfx1250 (MI455X), fp32 end-to-end via
// V_WMMA_F32_16X16X4_F32.
//   Phase 1: xw0[t][b][h] = x[b][t][:] @ W_ih_l0^T + (b_ih0+b_hh0)   (to d_ws)
//   Phase 2: sequential scan, 16 batch rows per wave32, WMMA matvecs.
// Workspace requirement: T*B*H*4 = 80 MiB (fits MI455X's 192 MB L2).
// ---------------------------------------------------------------------------

typedef __attribute__((ext_vector_type(2))) float v2f;
typedef __attribute__((ext_vector_type(8))) float v8f;

#define WMMA_F32_16x16x4(a, b, c) \
    __builtin_amdgcn_wmma_f32_16x16x4_f32(false, (a), false, (b), (short)0, (c), false, false)

constexpr int Bc = 8192;
constexpr int Tc = 256;
constexpr int Ic = 28;
constexpr int Hc = 10;

// ---------------------------------------------------------------------------
// Kernel 1: big input projection, one 16x(28)x16 WMMA tile per wave.
// Rows are flattened (b*T + t) indices over x viewed as [B*T, 28].
// Output layout: ws[t * B*H + b * H + h]  ([T,B,H], L2-resident).
// ---------------------------------------------------------------------------
__global__ __launch_bounds__(256)
void rnn_xproj_kernel(const float* __restrict__ x,
                      const float* __restrict__ W0,   // [10][28]
                      const float* __restrict__ bi0,  // [10]
                      const float* __restrict__ bh0,  // [10]
                      float* __restrict__ xw0)        // [T][B][H]
{
    const int lane  = threadIdx.x & 31;
    const int wave  = blockIdx.x * (blockDim.x >> 5) + (threadIdx.x >> 5);
    const int row0  = wave * 16;                  // first (b*T+t) row of tile
    if (row0 >= Bc * Tc) return;                  // wave-uniform

    const int n     = lane & 15;                  // B/C column (hidden idx)
    const int khalf = (lane >> 4) * 2;            // K offset per lane-half
    const int mrow  = lane & 15;                  // A row within tile
    const int mbase = (lane >> 4) * 8;            // C row base for this half

    // B operand: B[k][n] = W0[n][k], 7 chunks of K=4
    v2f bw[7];
#pragma unroll
    for (int c = 0; c < 7; ++c) {
        float w0 = 0.f, w1 = 0.f;
        if (n < Hc) {
            const int k = 4 * c + khalf;
            w0 = W0[n * Ic + k];
            w1 = W0[n * Ic + k + 1];
        }
        bw[c][0] = w0; bw[c][1] = w1;
    }

    const float bias = (n < Hc) ? (bi0[n] + bh0[n]) : 0.f;
    v8f acc;
#pragma unroll
    for (int r = 0; r < 8; ++r) acc[r] = bias;

    const float* xrow = x + (size_t)(row0 + mrow) * Ic;
#pragma unroll
    for (int c = 0; c < 7; ++c) {
        const int k = 4 * c + khalf;
        v2f a;
        a[0] = xrow[k];
        a[1] = xrow[k + 1];
        acc = WMMA_F32_16x16x4(a, bw[c], acc);
    }

    // Tile rows share one batch index b (T=256 is a multiple of 16).
    const int b  = row0 / Tc;
    const int t0 = row0 % Tc;
    if (n < Hc) {
#pragma unroll
        for (int r = 0; r < 8; ++r) {
            const int t = t0 + mbase + r;
            xw0[(size_t)t * (Bc * Hc) + (size_t)b * Hc + n] = acc[r];
        }
    }
}

// ---------------------------------------------------------------------------
// Kernel 2: sequential recurrence. One wave32 per 16 batch rows.
// Per step: 3 WMMAs (h0*Whh0^T, K pad 12) -> tanh -> LDS transpose ->
//           6 WMMAs (h0*Wih1^T + h1*Whh1^T) -> tanh -> store + LDS transpose.
// ---------------------------------------------------------------------------
__global__ __launch_bounds__(32)
void rnn_recur_kernel(const float* __restrict__ xw0,    // [T][B][H]
                      const float* __restrict__ h0init, // [2][B][H]
                      const float* __restrict__ Whh0,   // [10][10]
                      const float* __restrict__ Wih1,   // [10][10]
                      const float* __restrict__ Whh1,   // [10][10]
                      const float* __restrict__ bi1,
                      const float* __restrict__ bh1,
                      float* __restrict__ out)          // [B][T][H] ++ [2][B][H]
{
    __shared__ float lds0[16 * 16];
    __shared__ float lds1[16 * 16];

    const int lane  = threadIdx.x & 31;
    const int n     = lane & 15;
    const int khalf = (lane >> 4) * 2;
    const int mrow  = lane & 15;
    const int mbase = (lane >> 4) * 8;
    const int b0    = blockIdx.x * 16;

    // Load a 10x10 weight matrix into B-operand chunks: B[k][n] = W[n][k]
    auto loadW = [&](const float* __restrict__ W, v2f* dst) {
#pragma unroll
        for (int c = 0; c < 3; ++c) {
            const int k = 4 * c + khalf;
            float w0 = 0.f, w1 = 0.f;
            if (n < Hc) {
                if (k     < Hc) w0 = W[n * Hc + k];
                if (k + 1 < Hc) w1 = W[n * Hc + k + 1];
            }
            dst[c][0] = w0; dst[c][1] = w1;
        }
    };

    v2f bWhh0[3], bWih1[3], bWhh1[3];
    loadW(Whh0, bWhh0);
    loadW(Wih1, bWih1);
    loadW(Whh1, bWhh1);

    const float bias1 = (n < Hc) ? (bi1[n] + bh1[n]) : 0.f;

    // Initial hidden states as A-operand chunks (A[m][k] = h[b0+m][k])
    v2f a0[3], a1[3];
#pragma unroll
    for (int c = 0; c < 3; ++c) {
        const int k = 4 * c + khalf;
        float x0 = 0.f, x1 = 0.f, y0 = 0.f, y1 = 0.f;
        if (k < Hc) {
            x0 = h0init[(size_t)(b0 + mrow) * Hc + k];
            y0 = h0init[(size_t)(Bc + b0 + mrow) * Hc + k];
        }
        if (k + 1 < Hc) {
            x1 = h0init[(size_t)(b0 + mrow) * Hc + k + 1];
            y1 = h0init[(size_t)(Bc + b0 + mrow) * Hc + k + 1];
        }
        a0[c][0] = x0; a0[c][1] = x1;
        a1[c][0] = y0; a1[c][1] = y1;
    }

    v8f d0, d1;
#pragma unroll
    for (int r = 0; r < 8; ++r) { d0[r] = 0.f; d1[r] = 0.f; }

    for (int t = 0; t < Tc; ++t) {
        // ---- layer 0: c0 = xw0 tile; c0 += h0 @ Whh0^T; h0 = tanh(c0)
        const float* xwt = xw0 + (size_t)t * (Bc * Hc);
        v8f c0;
#pragma unroll
        for (int r = 0; r < 8; ++r)
            c0[r] = (n < Hc) ? xwt[(size_t)(b0 + mbase + r) * Hc + n] : 0.f;
#pragma unroll
        for (int c = 0; c < 3; ++c)
            c0 = WMMA_F32_16x16x4(a0[c], bWhh0[c], c0);
#pragma unroll
        for (int r = 0; r < 8; ++r) d0[r] = tanhf(c0[r]);

        // D layout -> LDS -> A layout (padding cols n>=10 are zero by constr.)
#pragma unroll
        for (int r = 0; r < 8; ++r) lds0[(mbase + r) * 16 + n] = d0[r];
        __syncthreads();
#pragma unroll
        for (int c = 0; c < 3; ++c) {
            const int k = 4 * c + khalf;
            a0[c][0] = lds0[mrow * 16 + k];
            a0[c][1] = lds0[mrow * 16 + k + 1];
        }

        // ---- layer 1: c1 = b1 + h0 @ Wih1^T + h1 @ Whh1^T; h1 = tanh(c1)
        v8f c1;
#pragma unroll
        for (int r = 0; r < 8; ++r) c1[r] = bias1;
#pragma unroll
        for (int c = 0; c < 3; ++c)
            c1 = WMMA_F32_16x16x4(a0[c], bWih1[c], c1);
#pragma unroll
        for (int c = 0; c < 3; ++c)
            c1 = WMMA_F32_16x16x4(a1[c], bWhh1[c], c1);
#pragma unroll
        for (int r = 0; r < 8; ++r) d1[r] = tanhf(c1[r]);

        // output[b][t][h]
        if (n < Hc) {
#pragma unroll
            for (int r = 0; r < 8; ++r)
                out[(size_t)(b0 + mbase + r) * (Tc * Hc) + (size_t)t * Hc + n] = d1[r];
        }

        // h1 D layout -> LDS -> A layout for next step
#pragma unroll
        for (int r = 0; r < 8; ++r) lds1[(mbase + r) * 16 + n] = d1[r];
        __syncthreads();
#pragma unroll
        for (int c = 0; c < 3; ++c) {
            const int k = 4 * c + khalf;
            a1[c][0] = lds1[mrow * 16 + k];
            a1[c][1] = lds1[mrow * 16 + k + 1];
        }
    }

    // h_final [2][B][H] appended after output
    float* hf = out + (size_t)Bc * Tc * Hc;
    if (n < Hc) {
#pragma unroll
        for (int r = 0; r < 8; ++r) {
            hf[(size_t)(b0 + mbase + r) * Hc + n]       = d0[r];
            hf[(size_t)(Bc + b0 + mbase + r) * Hc + n]  = d1[r];
        }
    }
}

// ---------------------------------------------------------------------------
extern "C" void kernel_launch(void* const* d_in, const int* in_sizes, int n_in,
                              void* d_out, int out_size, void* d_ws, size_t ws_size,
                              hipStream_t stream) {
    const float* x    = (const float*)d_in[0]; // [B,T,I]
    const float* h0   = (const float*)d_in[1]; // [2,B,H]
    const float* Wih0 = (const float*)d_in[2]; // [H,I]
    const float* Whh0 = (const float*)d_in[3]; // [H,H]
    const float* bih0 = (const float*)d_in[4];
    const float* bhh0 = (const float*)d_in[5];
    const float* Wih1 = (const float*)d_in[6]; // [H,H]
    const float* Whh1 = (const float*)d_in[7]; // [H,H]
    const float* bih1 = (const float*)d_in[8];
    const float* bhh1 = (const float*)d_in[9];
    float* out = (float*)d_out;
    float* xw0 = (float*)d_ws;                 // needs T*B*H*4 = 80 MiB

    // Phase 1: 131072 tiles, 8 waves (tiles) per 256-thread block
    const int tiles   = (Bc * Tc) / 16;
    const int blocks1 = tiles / 8;
    rnn_xproj_kernel<<<blocks1, 256, 0, stream>>>(x, Wih0, bih0, bhh0, xw0);

    // Phase 2: one wave per 16 batch rows
    rnn_recur_kernel<<<Bc / 16, 32, 0, stream>>>(xw0, h0, Whh0, Wih1, Whh1,
                                                 bih1, bhh1, out);
}